// PPEG_60705067762281
// MI455X (gfx1250) — compile-verified
//
#include <hip/hip_runtime.h>

// ---------------------------------------------------------------------------
// PPEG (depthwise 7x7 + 5x5 + 3x3 + identity, fused into one 7x7 filter),
// computed with V_WMMA_F32_16X16X4_F32 via banded-Toeplitz weight matrices.
// Input tile staged global->LDS with the Tensor Data Mover (TENSOR_LOAD_TO_LDS,
// iterate mode), overlapped with accumulator init; s_wait_tensorcnt to sync.
// ---------------------------------------------------------------------------

#define DIM   512
#define BATCH 16
#define HG    64
#define WG    64
#define NTOK  (1 + HG * WG)

typedef __attribute__((ext_vector_type(2))) float v2f;
typedef __attribute__((ext_vector_type(8))) float v8f;
typedef __attribute__((ext_vector_type(4))) unsigned int u32x4;
typedef __attribute__((ext_vector_type(8))) int i32x8;
typedef __attribute__((ext_vector_type(4))) int i32x4;

#if defined(__has_builtin)
#  if __has_builtin(__builtin_amdgcn_tensor_load_to_lds) && __has_builtin(__builtin_amdgcn_s_wait_tensorcnt)
#    define HAVE_TDM 1
#  endif
#endif

#define CH    8                  // channels per block == waves per block
#define ROWS  22                 // 16 output rows + 6 halo
#define RSF   580                // LDS row stride (floats): 72 cols * 8 ch + 4 (bank spread, RSF%64==4)
#define IN_FLOATS  (ROWS * RSF)             // 12760
#define WL_OFF     IN_FLOATS
#define BL_OFF     (WL_OFF + CH * 49)       // + 392
#define OS_OFF     (BL_OFF + CH)            // + 8
#define OS_FLOATS  (16 * 64 * CH)           // 8192
#define SMEM_FLOATS (OS_OFF + OS_FLOATS)    // 21352 floats = 85408 B

// --- Prologue: fuse w7 + pad(w5) + pad(w3) + identity; sum biases ----------
__global__ void ppeg_fuse_weights(const float* __restrict__ w7, const float* __restrict__ b7,
                                  const float* __restrict__ w5, const float* __restrict__ b5,
                                  const float* __restrict__ w3, const float* __restrict__ b3,
                                  float* __restrict__ wc, float* __restrict__ bc) {
    int c = blockIdx.x * blockDim.x + threadIdx.x;
    if (c >= DIM) return;
#pragma unroll
    for (int ky = 0; ky < 7; ++ky) {
#pragma unroll
        for (int kx = 0; kx < 7; ++kx) {
            float v = w7[c * 49 + ky * 7 + kx];
            if (ky >= 1 && ky <= 5 && kx >= 1 && kx <= 5)
                v += w5[c * 25 + (ky - 1) * 5 + (kx - 1)];
            if (ky >= 2 && ky <= 4 && kx >= 2 && kx <= 4)
                v += w3[c * 9 + (ky - 2) * 3 + (kx - 2)];
            if (ky == 3 && kx == 3) v += 1.0f;   // identity (residual) folded in
            wc[c * 49 + ky * 7 + kx] = v;
        }
    }
    bc[c] = b7[c] + b5[c] + b3[c];
}

// --- CLS token passthrough -------------------------------------------------
__global__ void ppeg_cls_copy(const float* __restrict__ x, float* __restrict__ out) {
    int i = blockIdx.x * 256 + threadIdx.x;      // 16*512 = 8192 threads
    int b = i >> 9;
    int c = i & 511;
    size_t off = (size_t)b * NTOK * DIM + c;
    out[off] = x[off];
}

// --- Main conv: one block = (batch, 16-row stripe, 8 channels) -------------
__global__ __launch_bounds__(256) void ppeg_conv(const float* __restrict__ x,
                                                 const float* __restrict__ wc,
                                                 const float* __restrict__ bc,
                                                 float* __restrict__ out) {
    extern __shared__ float smem[];
    float* inT  = smem;             // [ROWS][72 cols][CH] (+4 spread) interleaved tile
    float* wl   = smem + WL_OFF;    // [CH][49] fused weights
    float* bl   = smem + BL_OFF;    // [CH] fused bias
    float* outS = smem + OS_OFF;    // [16*64][CH] output staging

    const int cbase = blockIdx.x * CH;
    const int hbase = blockIdx.y * 16;
    const int b     = blockIdx.z;
    const int tid   = threadIdx.x;
    const size_t xbase = (size_t)b * NTOK * DIM + DIM;   // skip CLS token

    // valid (clamped) row range of the halo tile
    const int hlo   = (hbase - 3 < 0) ? 0 : hbase - 3;
    const int hhi   = (hbase + 19 > HG) ? HG : hbase + 19;
    const int nrows = hhi - hlo;
    const int r0    = hlo - (hbase - 3);   // first valid LDS tile row

    // ---- zero the tile (halo + K-pad stay zero), stage weights ----
    for (int i = tid; i < IN_FLOATS; i += 256) inT[i] = 0.0f;
    for (int i = tid; i < CH * 49; i += 256)
        wl[i] = wc[(cbase + i / 49) * 49 + (i % 49)];
    if (tid < CH) bl[tid] = bc[cbase + tid];
    __syncthreads();

#if defined(HAVE_TDM)
    // ---- Tensor Data Mover: DMA valid region rows into LDS (iterate mode) ----
    // per iteration: 2D tile of tile_dim0=8 ch (contiguous) x tile_dim1=64 cols
    // (stride DIM), then lds_addr += RSF, global_addr += WG*DIM.
    if (tid < 32) {   // wave 0 issues; EXEC ignored by TDM, one issue per wave
        const unsigned lds_base = (unsigned)(uintptr_t)inT
                                + (unsigned)((r0 * RSF + 3 * CH) * 4);
        const unsigned long long gaddr =
            (unsigned long long)(uintptr_t)(x + xbase + (size_t)hlo * WG * DIM + cbase);

        u32x4 g0 = { 1u,                                    // count=1 (valid D#)
                     lds_base,                              // lds_addr (bytes)
                     (unsigned)(gaddr & 0xFFFFFFFFu),       // global_addr lo
                     (unsigned)((gaddr >> 32) & 0x01FFFFFFu) | (2u << 30) }; // hi | type=2
        i32x8 g1 = { (int)((2u << 16) | (1u << 19)),  // data_size=4B, iterate_enable
                     (int)(8u << 16),                 // tensor_dim0 lo16 (=8) << 16
                     (int)(64u << 16),                // tensor_dim0 hi | tensor_dim1 lo16 (=64)
                     (int)(8u << 16),                 // tensor_dim1 hi | tile_dim0 (=8 ch)
                     (int)(64u),                      // tile_dim1 (=64 cols) | tile_dim2=0
                     (int)(DIM),                      // tensor_dim0_stride lo32 (=512)
                     0, 0 };                          // stride hi / tensor_dim1_stride
        i32x4 g2 = { 0,                               // tensor_dim2 (unused)
                     (int)(RSF),                      // lds_addr_increment (elements)
                     (int)(WG * DIM),                 // global_addr_increment (elements)
                     (int)((unsigned)(nrows - 1) << 16) };  // iterate_count
        i32x4 g3 = { 0, 0, 0, 0 };
#  if __clang_major__ >= 23
        i32x8 gx = { 0, 0, 0, 0, 0, 0, 0, 0 };
        __builtin_amdgcn_tensor_load_to_lds(g0, g1, g2, g3, gx, 0);
#  else
        __builtin_amdgcn_tensor_load_to_lds(g0, g1, g2, g3, 0);
#  endif
    }
#else
    // ---- fallback: coalesced VMEM load of the valid region ----
    for (int i = tid; i < nrows * 64 * CH; i += 256) {
        int c   = i & (CH - 1);
        int pos = i >> 3;
        int col = pos & 63;
        int row = pos >> 6;
        inT[(r0 + row) * RSF + (col + 3) * CH + c] =
            x[xbase + ((size_t)(hlo + row) * WG + col) * DIM + cbase + c];
    }
#endif

    const int lane = tid & 31;
    const int wv   = tid >> 5;        // wave id == local channel
    const int n    = lane & 15;       // N (=output col) and M (=A-row) lane index
    const int half = lane >> 4;
    const float bias = bl[wv];        // wl/bl ready (barrier above)

    // 4 accumulators: 16x16 output tiles along the 64-wide stripe
    v8f acc[4];
#pragma unroll
    for (int t = 0; t < 4; ++t)
#pragma unroll
        for (int v = 0; v < 8; ++v) acc[t][v] = bias;

#if defined(HAVE_TDM)
    if (tid < 32) __builtin_amdgcn_s_wait_tensorcnt(0);   // TDM done (wave 0)
#endif
    __syncthreads();                                       // tile visible to all

#pragma unroll
    for (int dy = 0; dy < 7; ++dy) {
        // ---- banded-Toeplitz B chunks for this dy (12 floats, live in VGPRs)
        // B 4x16 f32 layout: VGPR v, lanes 0-15 -> K=v, lanes 16-31 -> K=v+2
        // B[jj, w] = wt[dy, jj - w] when 0 <= jj-w <= 6 else 0  (jj = kc*4 + K)
        float Bv[12];
#pragma unroll
        for (int kc = 0; kc < 6; ++kc) {
#pragma unroll
            for (int v = 0; v < 2; ++v) {
                int jj  = kc * 4 + v + 2 * half;
                int dx  = jj - n;
                int dxc = dx < 0 ? 0 : (dx > 6 ? 6 : dx);
                float wt = wl[wv * 49 + dy * 7 + dxc];
                Bv[kc * 2 + v] = (dx >= 0 && dx <= 6) ? wt : 0.0f;
            }
        }
        // A 16x4 f32 layout: lane M = lane&15; VGPR v: K = v + 2*(lane>>4)
        // A[m, jj] = inT[(m+dy)*RSF + (wt4*16 + jj)*8 + wv]
        const float* rp = inT + (n + dy) * RSF + 16 * half + wv;
#pragma unroll
        for (int kc = 0; kc < 6; ++kc) {
            v2f bm = { Bv[kc * 2], Bv[kc * 2 + 1] };
#pragma unroll
            for (int wt4 = 0; wt4 < 4; ++wt4) {
                int o = (wt4 * 16 + kc * 4) * CH;
                v2f a = { rp[o], rp[o + CH] };     // ds_load_2addr_b32
                acc[wt4] = __builtin_amdgcn_wmma_f32_16x16x4_f32(
                              false, a, false, bm, (short)0, acc[wt4], false, false);
            }
        }
    }

    // D layout: VGPR v, lanes 0-15 -> M=v, lanes 16-31 -> M=v+8; N=lane&15
#pragma unroll
    for (int wt4 = 0; wt4 < 4; ++wt4)
#pragma unroll
        for (int v = 0; v < 8; ++v) {
            int m = v + 8 * half;
            outS[(m * 64 + wt4 * 16 + n) * CH + wv] = acc[wt4][v];
        }
    __syncthreads();

    // ---- coalesced writeback (channels contiguous in global) ----
    const size_t obase = (size_t)b * NTOK * DIM + DIM;
    for (int i = tid; i < 16 * 64 * CH; i += 256) {
        int c   = i & (CH - 1);
        int pos = i >> 3;
        int w   = pos & 63;
        int m   = pos >> 6;
        out[obase + ((size_t)(hbase + m) * WG + w) * DIM + cbase + c] = outS[pos * CH + c];
    }
}

extern "C" void kernel_launch(void* const* d_in, const int* in_sizes, int n_in,
                              void* d_out, int out_size, void* d_ws, size_t ws_size,
                              hipStream_t stream) {
    const float* x  = (const float*)d_in[0];
    const float* w7 = (const float*)d_in[1];
    const float* b7 = (const float*)d_in[2];
    const float* w5 = (const float*)d_in[3];
    const float* b5 = (const float*)d_in[4];
    const float* w3 = (const float*)d_in[5];
    const float* b3 = (const float*)d_in[6];
    float* out = (float*)d_out;

    float* wcomb = (float*)d_ws;           // 512*49 floats
    float* bcomb = wcomb + DIM * 49;       // 512 floats

    ppeg_fuse_weights<<<8, 64, 0, stream>>>(w7, b7, w5, b5, w3, b3, wcomb, bcomb);
    ppeg_cls_copy<<<(BATCH * DIM) / 256, 256, 0, stream>>>(x, out);

    dim3 grid(DIM / CH, HG / 16, BATCH);
    ppeg_conv<<<grid, 256, SMEM_FLOATS * sizeof(float), stream>>>(x, wcomb, bcomb, out);
}